// NetVLAD_42202348651243
// MI455X (gfx1250) — compile-verified
//
#include <hip/hip_runtime.h>
#include <math.h>

typedef float v2f __attribute__((ext_vector_type(2)));
typedef float v4f __attribute__((ext_vector_type(4)));
typedef float v8f __attribute__((ext_vector_type(8)));

constexpr int Bn = 32, Tn = 4096, Dn = 256, Kn = 64;
constexpr int TT    = 32;          // t-rows per subtile
constexpr int NSUB  = 8;           // subtiles per workgroup
constexpr int CHUNK = TT * NSUB;   // 256 t-rows per workgroup
constexpr int XS    = 260;         // padded LDS stride for x tile (bank-conflict free, 16B aligned)
constexpr int LS    = 68;          // padded LDS stride for logits tile

static __device__ __forceinline__ v8f wmma_f32(v2f a, v2f b, v8f c) {
  // V_WMMA_F32_16X16X4_F32 : D = A(16x4) * B(4x16) + C(16x16), fp32
  return __builtin_amdgcn_wmma_f32_16x16x4_f32(false, a, false, b, (short)0, c,
                                               false, false);
}

__global__ void zero_f32(float* __restrict__ p, int n) {
  int i = blockIdx.x * blockDim.x + threadIdx.x;
  if (i < n) p[i] = 0.f;
}

__global__ void __launch_bounds__(256)
netvlad_main(const float* __restrict__ x,      // (B,T,D)
             const float* __restrict__ w,      // (D,K)
             const float* __restrict__ bias,   // (K)
             float* __restrict__ vacc,         // (B,K,D) accumulator
             float* __restrict__ asum)         // (B,K) accumulator
{
  __shared__ float xs[TT * XS];   // x subtile, 32 x 256 (padded)
  __shared__ float ls[TT * LS];   // logits / soft-assign subtile, 32 x 64 (padded)

  const int tid  = threadIdx.x;
  const int lane = tid & 31;
  const int wave = tid >> 5;      // 8 waves
  const int hl   = lane >> 4;     // half-select (0/1)
  const int l16  = lane & 15;

  const int b  = blockIdx.x / (Tn / CHUNK);
  const int t0 = (blockIdx.x % (Tn / CHUNK)) * CHUNK;

  // Persistent phase-2 accumulators: each wave owns 8 output tiles of (K=64) x (D=256)
  v8f acc2[8] = {};
  float asum_part = 0.f;

  // Phase-1 tile assignment: 8 tiles (2 over t-rows x 4 over k-cols), one per wave
  const int mt = wave >> 2;       // 0..1
  const int nt = wave & 3;        // 0..3
  const float bk = bias[nt * 16 + l16];

  for (int sub = 0; sub < NSUB; ++sub) {
    // ---------------- load x subtile into LDS (float4, coalesced) ----------------
    const float* xsrc = x + ((size_t)b * Tn + (size_t)(t0 + sub * TT)) * Dn;
#pragma unroll
    for (int i = 0; i < (TT * (Dn / 4)) / 256; ++i) {
      int f   = tid + i * 256;
      int row = f >> 6;           // 0..31
      int col = (f & 63) << 2;    // 0..252
      *(v4f*)(xs + row * XS + col) = *(const v4f*)(xsrc + row * Dn + col);
    }
    __syncthreads();

    // ---------------- phase 1: logits = Xtile(32x256) @ W(256x64) ----------------
    {
      v8f c = {};
#pragma unroll 4
      for (int kk = 0; kk < Dn; kk += 4) {
        const int kA = kk + hl * 2;
        v2f a, bb;
        a.x  = xs[(mt * 16 + l16) * XS + kA];          // A: M=t-row, K=d
        a.y  = xs[(mt * 16 + l16) * XS + kA + 1];
        bb.x = w[kA * Kn + nt * 16 + l16];             // B: K=d, N=k
        bb.y = w[(kA + 1) * Kn + nt * 16 + l16];
        c = wmma_f32(a, bb, c);
      }
#pragma unroll
      for (int r = 0; r < 8; ++r)
        ls[(mt * 16 + hl * 8 + r) * LS + nt * 16 + l16] = c[r] + bk;
    }
    __syncthreads();

    // ---------------- softmax over k per t-row (8 lanes per row) ----------------
    {
      const int row = tid >> 3;   // 0..31
      const int seg = tid & 7;
      float* lr = ls + row * LS + seg * 8;
      float m = -1e30f;
      float e[8];
#pragma unroll
      for (int j = 0; j < 8; ++j) m = fmaxf(m, lr[j]);
      m = fmaxf(m, __shfl_xor(m, 1));
      m = fmaxf(m, __shfl_xor(m, 2));
      m = fmaxf(m, __shfl_xor(m, 4));
      float s = 0.f;
#pragma unroll
      for (int j = 0; j < 8; ++j) { e[j] = __expf(lr[j] - m); s += e[j]; }
      s += __shfl_xor(s, 1);
      s += __shfl_xor(s, 2);
      s += __shfl_xor(s, 4);
      const float inv = 1.f / s;
#pragma unroll
      for (int j = 0; j < 8; ++j) lr[j] = e[j] * inv;
    }
    __syncthreads();

    // ---------------- per-cluster assignment mass (column sums) ----------------
    {
      const int col = tid & 63;
      const int rb  = tid >> 6;   // 4 row-blocks of 8 rows
      float s = 0.f;
#pragma unroll
      for (int r = 0; r < 8; ++r) s += ls[(rb * 8 + r) * LS + col];
      asum_part += s;
    }

    // ---------------- phase 2: vlad += a^T(64 x 32) @ Xtile(32 x 256) ----------------
#pragma unroll
    for (int i = 0; i < 8; ++i) {
      const int ti = wave * 8 + i;   // 0..63 tiles: (K/16)=4 x (D/16)=16
      const int kt = ti >> 4;
      const int dt = ti & 15;
      v8f c = acc2[i];
#pragma unroll
      for (int tt = 0; tt < TT; tt += 4) {
        const int tA = tt + hl * 2;
        v2f a, bb;
        a.x  = ls[tA * LS + kt * 16 + l16];            // A = a^T: M=k, K=t
        a.y  = ls[(tA + 1) * LS + kt * 16 + l16];
        bb.x = xs[tA * XS + dt * 16 + l16];            // B = x: K=t, N=d
        bb.y = xs[(tA + 1) * XS + dt * 16 + l16];
        c = wmma_f32(a, bb, c);
      }
      acc2[i] = c;
    }
    __syncthreads();
  }

  // ---------------- commit accumulators ----------------
  unsafeAtomicAdd(&asum[b * Kn + (tid & 63)], asum_part);
#pragma unroll
  for (int i = 0; i < 8; ++i) {
    const int ti = wave * 8 + i;
    const int kt = ti >> 4;
    const int dt = ti & 15;
#pragma unroll
    for (int r = 0; r < 8; ++r) {
      const int kidx = kt * 16 + hl * 8 + r;           // D-layout: VGPR r -> M=r (+8 for hi lanes)
      const int d    = dt * 16 + l16;
      unsafeAtomicAdd(&vacc[((size_t)b * Kn + kidx) * Dn + d], acc2[i][r]);
    }
  }
}

__global__ void __launch_bounds__(256)
netvlad_finalize(const float* __restrict__ vacc, const float* __restrict__ asum,
                 const float* __restrict__ cent, float* __restrict__ out)
{
  const int bk   = blockIdx.x;        // b*64 + k
  const int k    = bk & 63;
  const int d    = threadIdx.x;       // 256 threads = one feature each
  const int lane = d & 31;
  const int wv   = d >> 5;
  __shared__ float red[8];

  float v = vacc[(size_t)bk * Dn + d] - asum[bk] * cent[k * Dn + d];
  float s = v * v;
#pragma unroll
  for (int off = 1; off < 32; off <<= 1) s += __shfl_xor(s, off);
  if (lane == 0) red[wv] = s;
  __syncthreads();
  float tot = 0.f;
#pragma unroll
  for (int i = 0; i < 8; ++i) tot += red[i];
  const float nrm = fmaxf(sqrtf(tot), 1e-12f);
  out[(size_t)bk * Dn + d] = v / nrm;
}

extern "C" void kernel_launch(void* const* d_in, const int* in_sizes, int n_in,
                              void* d_out, int out_size, void* d_ws, size_t ws_size,
                              hipStream_t stream) {
  const float* x    = (const float*)d_in[0];   // (B,T,D)
  const float* cent = (const float*)d_in[1];   // (K,D)
  const float* w    = (const float*)d_in[2];   // (D,K)
  const float* bias = (const float*)d_in[3];   // (K)
  float* out  = (float*)d_out;

  float* vacc = (float*)d_ws;                  // B*K*D floats
  float* asum = vacc + (size_t)Bn * Kn * Dn;   // B*K floats

  const int nzero = Bn * Kn * Dn + Bn * Kn;
  zero_f32<<<(nzero + 255) / 256, 256, 0, stream>>>(vacc, nzero);

  const int grid_main = Bn * (Tn / CHUNK);     // 32 * 16 = 512 workgroups
  netvlad_main<<<grid_main, 256, 0, stream>>>(x, w, bias, vacc, asum);

  netvlad_finalize<<<Bn * Kn, 256, 0, stream>>>(vacc, asum, cent, out);
}